// VRWKV_Rec_HWC_SpatialMix_12249246728669
// MI455X (gfx1250) — compile-verified
//
#include <hip/hip_runtime.h>
#include <math.h>

// ---------------------------------------------------------------------------
// VRWKV spatial-mix forward for MI455X (gfx1250, wave32, WMMA).
// GEMMs use V_WMMA_F32_16X16X4_F32 with 64x64 per-wave register blocking
// (16 WMMAs per K4 step from 8 b64 loads -> matrix-pipe bound, not L0 bound).
// WKV linear recurrences are chunk-parallelized (decay^L carry algebra).
// ---------------------------------------------------------------------------

typedef float v2f __attribute__((ext_vector_type(2)));
typedef float v8f __attribute__((ext_vector_type(8)));

#define C_DIM   768
#define H_DIM   64
#define W_DIM   64
#define T_DIM   (H_DIM * W_DIM)     // 4096
#define NCHUNK  16
#define LCHUNK  (T_DIM / NCHUNK)    // 256

// t -> physical position. transposed: logical t = x*h + y over original (y,x).
__device__ __forceinline__ int map_pos(int t, int transposed) {
    return transposed ? (((t & (H_DIM - 1)) << 6) + (t >> 6)) : t;
}

// ---------------------------------------------------------------------------
// 1) Depthwise conv blend: xs = a0*x + a1*dw1(x) + a2*dw3(x) + a3*dw5(x)
//    operating directly on (B,T,C) layout with spatial index offsets.
// ---------------------------------------------------------------------------
__global__ void conv_blend_kernel(const float* __restrict__ x,
                                  const float* __restrict__ w1,
                                  const float* __restrict__ w3,
                                  const float* __restrict__ w5,
                                  const float* __restrict__ alpha,
                                  float* __restrict__ out, int B)
{
    int tid = blockIdx.x * blockDim.x + threadIdx.x;
    int total = B * T_DIM * C_DIM;
    if (tid >= total) return;
    int c = tid % C_DIM;
    int t = (tid / C_DIM) % T_DIM;
    int b = tid / (C_DIM * T_DIM);
    int y = t / W_DIM, xx = t % W_DIM;

    float a0 = alpha[0], a1 = alpha[1], a2 = alpha[2], a3 = alpha[3];
    const float* base = x + (size_t)b * T_DIM * C_DIM;

    float ctr = base[(size_t)t * C_DIM + c];
    float acc = (a0 + a1 * w1[c]) * ctr;

    for (int dy = -2; dy <= 2; ++dy) {
        int yy = y + dy;
        if (yy < 0 || yy >= H_DIM) continue;
        for (int dx = -2; dx <= 2; ++dx) {
            int xs = xx + dx;
            if (xs < 0 || xs >= W_DIM) continue;
            float pv   = base[(size_t)(yy * W_DIM + xs) * C_DIM + c];
            float coef = a3 * w5[c * 25 + (dy + 2) * 5 + (dx + 2)];
            if (dy >= -1 && dy <= 1 && dx >= -1 && dx <= 1)
                coef += a2 * w3[c * 9 + (dy + 1) * 3 + (dx + 1)];
            acc = fmaf(coef, pv, acc);
        }
    }
    out[tid] = acc;
}

// ---------------------------------------------------------------------------
// 2) GEMM: D[M,N] = act( (A (⊙A2)) @ W^T + bias ), K = N = 768.
//    8 waves/block (4 along M, 2 along N); wave tile 64x64:
//      per K4 step: 4 A-frag + 4 B-frag b64 loads feed 16 WMMAs.
//    Block tile 256x128, grid (M/256, 6).
//    A-frag layout (16x4 f32): lanes 0-15 hold K={k0,k0+1}, lanes 16-31
//    hold K={k0+2,k0+3}; B mirrors with N striped over lanes.
// ---------------------------------------------------------------------------
template<bool GATE, bool SIG>
__global__ __launch_bounds__(256)
void gemm_wmma_f32(const float* __restrict__ A, const float* __restrict__ A2,
                   const float* __restrict__ W, const float* __restrict__ bias,
                   float* __restrict__ D, int M)
{
    const int K = C_DIM, N = C_DIM;
    int lane  = threadIdx.x & 31;
    int wave  = threadIdx.x >> 5;
    int waveM = wave & 3, waveN = wave >> 2;
    int m0 = blockIdx.x * 256 + waveM * 64;
    int n0 = blockIdx.y * 128 + waveN * 64;
    int r  = lane & 15, hi = lane >> 4;

    v8f acc[4][4]; // [mi][ni]
    #pragma unroll
    for (int mi = 0; mi < 4; ++mi)
        #pragma unroll
        for (int ni = 0; ni < 4; ++ni) {
            v8f z = {0.f, 0.f, 0.f, 0.f, 0.f, 0.f, 0.f, 0.f};
            acc[mi][ni] = z;
        }

    const float* aP[4];
    const float* gP[4];
    const float* wP[4];
    #pragma unroll
    for (int mi = 0; mi < 4; ++mi) {
        aP[mi] = A + (size_t)(m0 + 16 * mi + r) * K + 2 * hi;
        if (GATE) gP[mi] = A2 + (size_t)(m0 + 16 * mi + r) * K + 2 * hi;
    }
    #pragma unroll
    for (int ni = 0; ni < 4; ++ni)
        wP[ni] = W + (size_t)(n0 + 16 * ni + r) * K + 2 * hi;

    for (int kb = 0; kb < K; kb += 64) {
        if (kb + 64 < K) {
            #pragma unroll
            for (int mi = 0; mi < 4; ++mi) __builtin_prefetch(aP[mi] + kb + 64, 0, 3);
            #pragma unroll
            for (int ni = 0; ni < 4; ++ni) __builtin_prefetch(wP[ni] + kb + 64, 0, 3);
        }
        #pragma unroll
        for (int k4 = 0; k4 < 64; k4 += 4) {
            int k0 = kb + k4;
            v2f af[4], bf[4];
            #pragma unroll
            for (int mi = 0; mi < 4; ++mi) {
                af[mi] = *(const v2f*)(aP[mi] + k0);
                if (GATE) {
                    v2f g = *(const v2f*)(gP[mi] + k0);
                    af[mi].x *= g.x; af[mi].y *= g.y;
                }
            }
            #pragma unroll
            for (int ni = 0; ni < 4; ++ni)
                bf[ni] = *(const v2f*)(wP[ni] + k0);
            #pragma unroll
            for (int mi = 0; mi < 4; ++mi)
                #pragma unroll
                for (int ni = 0; ni < 4; ++ni)
                    acc[mi][ni] = __builtin_amdgcn_wmma_f32_16x16x4_f32(
                        false, af[mi], false, bf[ni], (short)0, acc[mi][ni],
                        false, false);
        }
    }

    // C/D layout: lanes 0-15 -> N=col band + r with M rows 0..7;
    // lanes 16-31 -> same cols, M rows 8..15.
    #pragma unroll
    for (int mi = 0; mi < 4; ++mi) {
        #pragma unroll
        for (int ni = 0; ni < 4; ++ni) {
            int col = n0 + 16 * ni + r;
            float bv = bias ? bias[col] : 0.f;
            #pragma unroll
            for (int i = 0; i < 8; ++i) {
                int row = m0 + 16 * mi + hi * 8 + i;
                float val = acc[mi][ni][i] + bv;
                if (SIG) val = 1.f / (1.f + __expf(-val));
                D[(size_t)row * N + col] = val;
            }
        }
    }
}

// ---------------------------------------------------------------------------
// 3) WKV chunked scan. ew = exp(-exp(dec/T)), u = first/T.
//    a_f[t] = sum_{s<t} ew^{t-1-s} e^{k}v ;  a_r[t] = sum_{s>t} ew^{s-t-1} e^{k}v
//    out = (a_f + a_r + e^{u+k} v) / (b_f + b_r + e^{u+k})
//    Chunk carry: carry_out = ew^L * carry_in + local_sum.
// ---------------------------------------------------------------------------
__global__ void wkv_local(const float* __restrict__ k, const float* __restrict__ v,
                          const float* __restrict__ dec,
                          float* __restrict__ scrQ, int B, int transposed)
{
    int tid = blockIdx.x * blockDim.x + threadIdx.x;
    int total = B * NCHUNK * C_DIM;
    if (tid >= total) return;
    int c = tid % C_DIM;
    int j = (tid / C_DIM) % NCHUNK;
    int b = tid / (C_DIM * NCHUNK);

    float ew = __expf(-__expf(dec[c] * (1.f / T_DIM)));
    size_t basebc = (size_t)b * T_DIM * C_DIM + c;

    float fa = 0.f, fb = 0.f;
    for (int t = j * LCHUNK; t < (j + 1) * LCHUNK; ++t) {
        size_t idx = basebc + (size_t)map_pos(t, transposed) * C_DIM;
        float kk = k[idx], vv = v[idx];
        float ek = __expf(kk);
        fa = fmaf(ew, fa, ek * vv);
        fb = fmaf(ew, fb, ek);
    }
    float ra = 0.f, rb = 0.f;
    for (int t = (j + 1) * LCHUNK - 1; t >= j * LCHUNK; --t) {
        size_t idx = basebc + (size_t)map_pos(t, transposed) * C_DIM;
        float kk = k[idx], vv = v[idx];
        float ek = __expf(kk);
        ra = fmaf(ew, ra, ek * vv);
        rb = fmaf(ew, rb, ek);
    }
    size_t si    = ((size_t)b * NCHUNK + j) * C_DIM + c;
    size_t plane = (size_t)B * NCHUNK * C_DIM;
    scrQ[si]             = fa;
    scrQ[plane + si]     = fb;
    scrQ[2 * plane + si] = ra;
    scrQ[3 * plane + si] = rb;
}

__global__ void wkv_carry(const float* __restrict__ dec,
                          const float* __restrict__ scrQ,
                          float* __restrict__ scrC, int B)
{
    int tid = blockIdx.x * blockDim.x + threadIdx.x;
    if (tid >= B * C_DIM) return;
    int c = tid % C_DIM;
    int b = tid / C_DIM;
    float ewL = __expf(-(float)LCHUNK * __expf(dec[c] * (1.f / T_DIM)));
    size_t plane = (size_t)B * NCHUNK * C_DIM;

    float a = 0.f, bb = 0.f;
    for (int j = 0; j < NCHUNK; ++j) {
        size_t si = ((size_t)b * NCHUNK + j) * C_DIM + c;
        scrC[si]         = a;   // forward carry entering chunk j
        scrC[plane + si] = bb;
        a  = fmaf(ewL, a,  scrQ[si]);
        bb = fmaf(ewL, bb, scrQ[plane + si]);
    }
    float ra = 0.f, rb = 0.f;
    for (int j = NCHUNK - 1; j >= 0; --j) {
        size_t si = ((size_t)b * NCHUNK + j) * C_DIM + c;
        scrC[2 * plane + si] = ra;  // reverse carry entering chunk j (from right)
        scrC[3 * plane + si] = rb;
        ra = fmaf(ewL, ra, scrQ[2 * plane + si]);
        rb = fmaf(ewL, rb, scrQ[3 * plane + si]);
    }
}

// out doubles as den-scratch (same index touched only by this thread).
__global__ void wkv_combine(const float* __restrict__ k, const float* __restrict__ v,
                            float* __restrict__ out, float* __restrict__ numscr,
                            const float* __restrict__ dec, const float* __restrict__ frst,
                            const float* __restrict__ scrC, int B, int transposed)
{
    int tid = blockIdx.x * blockDim.x + threadIdx.x;
    int total = B * NCHUNK * C_DIM;
    if (tid >= total) return;
    int c = tid % C_DIM;
    int j = (tid / C_DIM) % NCHUNK;
    int b = tid / (C_DIM * NCHUNK);

    float invT = 1.f / T_DIM;
    float ew = __expf(-__expf(dec[c] * invT));
    float u  = frst[c] * invT;
    size_t basebc = (size_t)b * T_DIM * C_DIM + c;
    size_t si    = ((size_t)b * NCHUNK + j) * C_DIM + c;
    size_t plane = (size_t)B * NCHUNK * C_DIM;

    // descending: emit reverse part + e^{u+k} terms
    float ra = scrC[2 * plane + si], rb = scrC[3 * plane + si];
    for (int t = (j + 1) * LCHUNK - 1; t >= j * LCHUNK; --t) {
        size_t idx = basebc + (size_t)map_pos(t, transposed) * C_DIM;
        float kk = k[idx], vv = v[idx];
        float ek = __expf(kk);
        float eu = __expf(u + kk);
        numscr[idx] = fmaf(eu, vv, ra);
        out[idx]    = rb + eu;
        ra = fmaf(ew, ra, ek * vv);
        rb = fmaf(ew, rb, ek);
    }
    // ascending: add forward part, divide, write result
    float fa = scrC[si], fb = scrC[plane + si];
    for (int t = j * LCHUNK; t < (j + 1) * LCHUNK; ++t) {
        size_t idx = basebc + (size_t)map_pos(t, transposed) * C_DIM;
        float kk = k[idx], vv = v[idx];
        float ek = __expf(kk);
        float num = fa + numscr[idx];
        float den = fb + out[idx];
        out[idx] = num / den;
        fa = fmaf(ew, fa, ek * vv);
        fb = fmaf(ew, fb, ek);
    }
}

// ---------------------------------------------------------------------------
static void run_wkv(const float* k, const float* v, float* out, float* numscr,
                    const float* decrow, const float* frstrow,
                    float* scrQ, float* scrC, int B, int transposed,
                    hipStream_t stream)
{
    int nloc = B * NCHUNK * C_DIM;
    wkv_local<<<(nloc + 255) / 256, 256, 0, stream>>>(k, v, decrow, scrQ, B, transposed);
    int ncar = B * C_DIM;
    wkv_carry<<<(ncar + 255) / 256, 256, 0, stream>>>(decrow, scrQ, scrC, B);
    wkv_combine<<<(nloc + 255) / 256, 256, 0, stream>>>(k, v, out, numscr, decrow,
                                                        frstrow, scrC, B, transposed);
}

extern "C" void kernel_launch(void* const* d_in, const int* in_sizes, int n_in,
                              void* d_out, int out_size, void* d_ws, size_t ws_size,
                              hipStream_t stream)
{
    const float* x        = (const float*)d_in[0];
    const float* os_w1    = (const float*)d_in[3];
    const float* os_w3    = (const float*)d_in[4];
    const float* os_w5    = (const float*)d_in[5];
    const float* os_alpha = (const float*)d_in[6];
    const float* Wk       = (const float*)d_in[7];
    const float* Wv       = (const float*)d_in[8];
    const float* Wr       = (const float*)d_in[9];
    const float* Wf       = (const float*)d_in[10];
    const float* bfw      = (const float*)d_in[11];
    const float* Wb       = (const float*)d_in[12];
    const float* bbw      = (const float*)d_in[13];
    const float* Wo       = (const float*)d_in[14];
    const float* sdec     = (const float*)d_in[15];
    const float* sfirst   = (const float*)d_in[16];

    int B = in_sizes[0] / (T_DIM * C_DIM);
    int M = B * T_DIM;
    size_t bufElems = (size_t)M * C_DIM;

    float* X    = (float*)d_ws;          // xf, later reused as v ping buffer
    float* K0   = X  + bufElems;
    float* V0   = K0 + bufElems;
    float* SR   = V0 + bufElems;
    float* K1   = SR + bufElems;
    float* scrQ = K1 + bufElems;                         // 4 planes
    float* scrC = scrQ + (size_t)4 * B * NCHUNK * C_DIM; // 4 planes

    int total = M * C_DIM;
    conv_blend_kernel<<<(total + 255) / 256, 256, 0, stream>>>(
        x, os_w1, os_w3, os_w5, os_alpha, X, B);

    dim3 ggrid(M / 256, C_DIM / 128);
    // projections
    gemm_wmma_f32<false,false><<<ggrid, 256, 0, stream>>>(X, nullptr, Wk, nullptr, K0, M);
    gemm_wmma_f32<false,false><<<ggrid, 256, 0, stream>>>(X, nullptr, Wv, nullptr, V0, M);
    gemm_wmma_f32<false,true ><<<ggrid, 256, 0, stream>>>(X, nullptr, Wr, nullptr, SR, M);

    // j = 0: forward transform + wkv
    gemm_wmma_f32<false,false><<<ggrid, 256, 0, stream>>>(K0, nullptr, Wf, bfw, K1, M);
    gemm_wmma_f32<false,false><<<ggrid, 256, 0, stream>>>(V0, nullptr, Wf, bfw, X,  M);
    run_wkv(K1, X, V0, K0, sdec + 0 * C_DIM, sfirst + 0 * C_DIM, scrQ, scrC, B, 0, stream);

    // j = 1: transposed wkv on (k=K1, v=V0) -> X
    run_wkv(K1, V0, X, K0, sdec + 1 * C_DIM, sfirst + 1 * C_DIM, scrQ, scrC, B, 1, stream);

    // j = 2: backward transform + wkv
    gemm_wmma_f32<false,false><<<ggrid, 256, 0, stream>>>(K1, nullptr, Wb, bbw, K0, M);
    gemm_wmma_f32<false,false><<<ggrid, 256, 0, stream>>>(X,  nullptr, Wb, bbw, V0, M);
    run_wkv(K0, V0, X, K1, sdec + 2 * C_DIM, sfirst + 2 * C_DIM, scrQ, scrC, B, 0, stream);

    // j = 3: transposed wkv on (k=K0, v=X) -> V0
    run_wkv(K0, X, V0, K1, sdec + 3 * C_DIM, sfirst + 3 * C_DIM, scrQ, scrC, B, 1, stream);

    // out = (sr ⊙ v) @ Wo^T  (elementwise gate fused into A-load)
    gemm_wmma_f32<true,false><<<ggrid, 256, 0, stream>>>(V0, SR, Wo, nullptr, (float*)d_out, M);
}